// SkipGramNeg_1357209666155
// MI455X (gfx1250) — compile-verified
//
#include <hip/hip_runtime.h>

typedef __attribute__((ext_vector_type(2))) float v2f;
typedef __attribute__((ext_vector_type(8))) float v8f;
typedef __attribute__((__vector_size__(4 * sizeof(int)))) int i32x4;

#define ROWS   16
#define DIM    128
#define STRIDE 132   // DIM + 4 floats pad: breaks LDS bank conflicts, keeps 16B align

#define AS1 __attribute__((address_space(1)))
#define AS3 __attribute__((address_space(3)))

#if defined(__has_builtin)
#if __has_builtin(__builtin_amdgcn_global_load_async_to_lds_b128)
#define SG_HAS_ASYNC_LDS 1
#endif
#endif

__global__ void sg_zero_kernel(float* out) { out[0] = 0.0f; }

__device__ __forceinline__ float logsig(float x) {
    // numerically stable log(sigmoid(x)) = min(x,0) - log1p(exp(-|x|))
    return fminf(x, 0.0f) - log1pf(expf(-fabsf(x)));
}

__global__ __launch_bounds__(32)
void sg_loss_kernel(const int* __restrict__ tgt, const int* __restrict__ ctx,
                    const int* __restrict__ neg, const float* __restrict__ emb,
                    float* __restrict__ out, int B, int K, float scale)
{
    __shared__ float sV[ROWS * STRIDE];   // v rows       (16 x 128 f32)
    __shared__ float sU[ROWS * STRIDE];   // u rows
    __shared__ float sN[ROWS * STRIDE];   // sum over K negative rows

    const int lane = threadIdx.x;             // 0..31, one wave per block
    const int base = blockIdx.x * ROWS;       // first batch row of this group

    // ---- Stage 1: gather embedding rows into LDS ----
    // V and U rows go straight to LDS via the CDNA5 async DMA path (no VGPR
    // round-trip, tracked by ASYNCcnt); the K negative rows must pass through
    // VGPRs for the accumulation, so they use regular coalesced b128 loads.
    for (int m = 0; m < ROWS; ++m) {
        int b  = base + m;
        int bm = (b < B) ? b : 0;             // safe clamp; masked out later
        const float4* vp = (const float4*)(emb + (long)tgt[bm] * DIM);
        const float4* up = (const float4*)(emb + (long)ctx[bm] * DIM);
#if SG_HAS_ASYNC_LDS
        __builtin_amdgcn_global_load_async_to_lds_b128(
            (AS1 i32x4*)(vp + lane), (AS3 i32x4*)&sV[m * STRIDE + lane * 4], 0, 0);
        __builtin_amdgcn_global_load_async_to_lds_b128(
            (AS1 i32x4*)(up + lane), (AS3 i32x4*)&sU[m * STRIDE + lane * 4], 0, 0);
#else
        float4 v4 = vp[lane];                 // 32 lanes x 16B = full 512B row
        float4 u4 = up[lane];
        *(float4*)&sV[m * STRIDE + lane * 4] = v4;
        *(float4*)&sU[m * STRIDE + lane * 4] = u4;
#endif
        float4 a4 = make_float4(0.f, 0.f, 0.f, 0.f);
        const int* nrow = neg + (long)bm * K;
        #pragma unroll 4
        for (int k = 0; k < K; ++k) {
            const float4* np = (const float4*)(emb + (long)nrow[k] * DIM);
            float4 n4 = np[lane];
            a4.x += n4.x; a4.y += n4.y; a4.z += n4.z; a4.w += n4.w;
        }
        *(float4*)&sN[m * STRIDE + lane * 4] = a4;
    }
#if SG_HAS_ASYNC_LDS
#if __has_builtin(__builtin_amdgcn_s_wait_asynccnt)
    __builtin_amdgcn_s_wait_asynccnt(0);
#else
    asm volatile("s_wait_asynccnt 0" ::: "memory");
#endif
#endif
    __syncthreads();

    // ---- Stage 2: C_pos = V*U^T, C_neg = V*NS^T via V_WMMA_F32_16X16X4_F32 ----
    // Fragment layout (ISA 7.12.2, 32-bit 16x4 A / 4x16 B): lane l holds
    // row = l&15, k = 4*s + 2*(l>>4) + {0,1}; identical addressing for A and B.
    const int r    = lane & 15;
    const int koff = (lane >> 4) * 2;
    v8f cp = {};  // C/D accumulator 16x16 f32 = 8 VGPRs
    v8f cn = {};
    #pragma unroll
    for (int s = 0; s < DIM / 4; ++s) {
        const int k = s * 4 + koff;
        v2f a  = *(const v2f*)&sV[r * STRIDE + k];
        v2f bu = *(const v2f*)&sU[r * STRIDE + k];
        v2f bn = *(const v2f*)&sN[r * STRIDE + k];
        cp = __builtin_amdgcn_wmma_f32_16x16x4_f32(false, a, false, bu,
                                                   (short)0, cp, false, false);
        cn = __builtin_amdgcn_wmma_f32_16x16x4_f32(false, a, false, bn,
                                                   (short)0, cn, false, false);
    }

    // ---- Stage 3: extract diagonals, log-sigmoid, reduce ----
    // diag[m=j]   lives in VGPR j at lane j      (M=j,   N=lane)
    // diag[m=j+8] lives in VGPR j at lane j+24   (M=j+8, N=lane-16)
    float total = 0.0f;
    #pragma unroll
    for (int j = 0; j < 8; ++j) {
        float p1 = __shfl(cp[j], j,      32);
        float n1 = __shfl(cn[j], j,      32);
        float p2 = __shfl(cp[j], j + 24, 32);
        float n2 = __shfl(cn[j], j + 24, 32);
        if (base + j     < B) total += logsig(p1) + logsig(-n1);
        if (base + j + 8 < B) total += logsig(p2) + logsig(-n2);
    }
    if (lane == 0) atomicAdd(out, total * scale);
}

extern "C" void kernel_launch(void* const* d_in, const int* in_sizes, int n_in,
                              void* d_out, int out_size, void* d_ws, size_t ws_size,
                              hipStream_t stream) {
    const int*   tgt = (const int*)d_in[0];
    const int*   ctx = (const int*)d_in[1];
    const int*   neg = (const int*)d_in[2];
    const float* emb = (const float*)d_in[3];
    float*       out = (float*)d_out;

    const int B = in_sizes[0];
    const int K = in_sizes[2] / B;            // 20
    const float scale = -1.0f / (float)B;     // -mean(...)

    sg_zero_kernel<<<1, 1, 0, stream>>>(out);
    const int groups = (B + ROWS - 1) / ROWS; // 1024 waves for B=16384
    sg_loss_kernel<<<groups, 32, 0, stream>>>(tgt, ctx, neg, emb, out, B, K, scale);
}